// SimplePNA_9208409883076
// MI455X (gfx1250) — compile-verified
//
#include <hip/hip_runtime.h>
#include <math.h>

#define NODES   50000
#define EDGES   400000
#define DIM     128
#define TWO_DIM 256
#define U_IN    1664   // (4 aggregators * 3 scalers + 1) * 128
#define DELTA   4.0f

typedef __attribute__((ext_vector_type(2))) float v2f;
typedef __attribute__((ext_vector_type(8))) float v8f;

// CDNA5 fp32 matrix op: D(16x16) = A(16x4) * B(4x16) + C, wave32 striped.
__device__ __forceinline__ v8f wmma4(v2f a, v2f b, v8f c) {
  return __builtin_amdgcn_wmma_f32_16x16x4_f32(
      /*neg_a=*/false, a, /*neg_b=*/false, b,
      /*c_mod=*/(short)0, c, /*reuse_a=*/false, /*reuse_b=*/false);
}

// Float atomic max/min via monotonic int-bit trick (global_atomic_max_i32 etc.)
__device__ __forceinline__ void atomic_max_f32(float* a, float v) {
  if (v >= 0.f) atomicMax((int*)a, __float_as_int(v));
  else          atomicMin((unsigned int*)a, __float_as_uint(v));
}
__device__ __forceinline__ void atomic_min_f32(float* a, float v) {
  if (v >= 0.f) atomicMin((int*)a, __float_as_int(v));
  else          atomicMax((unsigned int*)a, __float_as_uint(v));
}

// ---------------------------------------------------------------- init
__global__ void __launch_bounds__(256) init_accum(
    float* __restrict__ s1, float* __restrict__ s2,
    float* __restrict__ mx, float* __restrict__ mn, float* __restrict__ deg) {
  int i = blockIdx.x * blockDim.x + threadIdx.x;
  if (i < NODES * DIM) {
    s1[i] = 0.f;
    s2[i] = 0.f;
    mx[i] = -__builtin_inff();
    mn[i] =  __builtin_inff();
  }
  if (i < NODES) deg[i] = 0.f;
}

__global__ void __launch_bounds__(256) deg_kernel(
    const int* __restrict__ dst, float* __restrict__ deg) {
  int i = blockIdx.x * blockDim.x + threadIdx.x;
  if (i >= EDGES + NODES) return;
  int d = (i < EDGES) ? dst[i] : (i - EDGES);   // implicit self loops
  atomicAdd(deg + d, 1.0f);
}

// ---------------------------------------------------------------- msg GEMM + scatter
// One wave per 32-edge tile (2 M-tiles share B-fragments):
//   msg(32x128) = [h[src],h[dst]](32x256) @ Mw^T, then scatter to s1/s2/mx/mn.
__global__ void __launch_bounds__(256) msg_kernel(
    const float* __restrict__ h, const int* __restrict__ src, const int* __restrict__ dst,
    const float* __restrict__ Mw, const float* __restrict__ Mb,
    float* __restrict__ s1, float* __restrict__ s2,
    float* __restrict__ mx, float* __restrict__ mn) {
  const int wave = (blockIdx.x * blockDim.x + threadIdx.x) >> 5;
  const int lane = threadIdx.x & 31;
  const int half = lane >> 4;
  const int lq   = lane & 15;
  const int Et   = EDGES + NODES;
  const int e0   = wave * 32;
  if (e0 >= Et) return;                 // wave-uniform: EXEC stays all-ones
  const bool t1v = (e0 + 16) < Et;      // wave-uniform second-tile validity

  // A-fragment source rows for this lane (lanes lq / lq+16 share row lq, K phases 0/2)
  const int ea0 = e0 + lq;
  const int ea1 = t1v ? (e0 + 16 + lq) : ea0;   // clamp loads when tile1 invalid
  const int sa0 = (ea0 < EDGES) ? src[ea0] : (ea0 - EDGES);
  const int da0 = (ea0 < EDGES) ? dst[ea0] : (ea0 - EDGES);
  const int sa1 = (ea1 < EDGES) ? src[ea1] : (ea1 - EDGES);
  const int da1 = (ea1 < EDGES) ? dst[ea1] : (ea1 - EDGES);
  const float* __restrict__ rowS0 = h + (size_t)sa0 * DIM;
  const float* __restrict__ rowD0 = h + (size_t)da0 * DIM;
  const float* __restrict__ rowS1 = h + (size_t)sa1 * DIM;
  const float* __restrict__ rowD1 = h + (size_t)da1 * DIM;

  v8f acc0[8], acc1[8];
#pragma unroll
  for (int n = 0; n < 8; ++n)
#pragma unroll
    for (int r = 0; r < 8; ++r) { acc0[n][r] = 0.f; acc1[n][r] = 0.f; }

  const int ko = half * 2;
  for (int ks = 0; ks < TWO_DIM / 4; ++ks) {
    const int k = ks * 4 + ko;
    v2f a0 = (k < DIM) ? *(const v2f*)(rowS0 + k) : *(const v2f*)(rowD0 + (k - DIM));
    v2f a1 = (k < DIM) ? *(const v2f*)(rowS1 + k) : *(const v2f*)(rowD1 + (k - DIM));
#pragma unroll
    for (int n = 0; n < 8; ++n) {
      const int nf = n * 16 + lq;       // B: lanes = N, vgprs = K rows
      v2f b = *(const v2f*)(Mw + (size_t)nf * TWO_DIM + k);
      acc0[n] = wmma4(a0, b, acc0[n]);  // B-fragment reused across both M-tiles
      acc1[n] = wmma4(a1, b, acc1[n]);
    }
  }

  // rows covered by this lane's accumulator registers: M = r + 8*half
  int dr0[8], dr1[8];
#pragma unroll
  for (int r = 0; r < 8; ++r) {
    const int ei0 = e0 + r + half * 8;
    dr0[r] = (ei0 < EDGES) ? dst[ei0] : (ei0 - EDGES);
    const int ei1 = e0 + 16 + r + half * 8;
    dr1[r] = t1v ? ((ei1 < EDGES) ? dst[ei1] : (ei1 - EDGES)) : 0;
  }

#pragma unroll
  for (int n = 0; n < 8; ++n) {
    const int f = n * 16 + lq;
    const float bias = Mb[f];
#pragma unroll
    for (int r = 0; r < 8; ++r) {
      {
        const float v = acc0[n][r] + bias;
        const size_t idx = (size_t)dr0[r] * DIM + f;
        atomicAdd(s1 + idx, v);
        atomicAdd(s2 + idx, v * v);
        atomic_max_f32(mx + idx, v);
        atomic_min_f32(mn + idx, v);
      }
      if (t1v) {
        const float v = acc1[n][r] + bias;
        const size_t idx = (size_t)dr1[r] * DIM + f;
        atomicAdd(s1 + idx, v);
        atomicAdd(s2 + idx, v * v);
        atomic_max_f32(mx + idx, v);
        atomic_min_f32(mn + idx, v);
      }
    }
  }
}

// ---------------------------------------------------------------- finalize: s1 -> std, s2 -> var (in place)
__global__ void __launch_bounds__(256) finalize_agg(
    float* __restrict__ s1, float* __restrict__ s2, const float* __restrict__ deg) {
  int i = blockIdx.x * blockDim.x + threadIdx.x;
  if (i >= NODES * DIM) return;
  const float dg  = deg[i >> 7];
  const float m   = s1[i] / dg;
  const float var = fmaxf(s2[i] / dg - m * m, 0.f);
  s1[i] = sqrtf(var + 1e-30f);
  s2[i] = var;
}

// ---------------------------------------------------------------- U GEMM (K = 1664, A built on the fly)
__device__ __forceinline__ v2f fetch_xu(
    const float* __restrict__ h,  const float* __restrict__ mx, const float* __restrict__ mn,
    const float* __restrict__ st, const float* __restrict__ vr,
    int rowbase, int k, float sc1, float sc2) {
  if (k < DIM) return *(const v2f*)(h + rowbase + k);
  const int kk    = k - DIM;
  const int reg   = kk >> 9;        // scaler region: 1, logd/delta, delta/logd
  const int inner = kk & 511;
  const int j     = inner >> 7;     // 0:max 1:min 2:std 3:var
  const int f     = inner & 127;
  const float* __restrict__ base = (j == 0) ? mx : (j == 1) ? mn : (j == 2) ? st : vr;
  v2f v = *(const v2f*)(base + rowbase + f);
  const float sc = (reg == 0) ? 1.0f : (reg == 1) ? sc1 : sc2;
  v.x *= sc; v.y *= sc;
  return v;
}

// One wave per 32-node tile (2 M-tiles share Uw B-fragments).
__global__ void __launch_bounds__(256) u_gemm_kernel(
    const float* __restrict__ h,  const float* __restrict__ mx, const float* __restrict__ mn,
    const float* __restrict__ st, const float* __restrict__ vr, const float* __restrict__ deg,
    const float* __restrict__ Uw, const float* __restrict__ Ub, float* __restrict__ u) {
  const int wave = (blockIdx.x * blockDim.x + threadIdx.x) >> 5;
  const int lane = threadIdx.x & 31;
  const int half = lane >> 4;
  const int lq   = lane & 15;
  const int n0   = wave * 32;
  if (n0 >= NODES) return;
  const bool t1v = (n0 + 16) < NODES;

  const int rowA0 = n0 + lq;
  const int rowA1 = t1v ? (n0 + 16 + lq) : rowA0;
  const int rb0   = rowA0 * DIM;
  const int rb1   = rowA1 * DIM;
  const float logd0 = logf(deg[rowA0] + 1.f);
  const float logd1 = logf(deg[rowA1] + 1.f);
  const float sc1a = logd0 * (1.0f / DELTA), sc2a = DELTA / logd0;
  const float sc1b = logd1 * (1.0f / DELTA), sc2b = DELTA / logd1;

  v8f acc0[8], acc1[8];
#pragma unroll
  for (int n = 0; n < 8; ++n)
#pragma unroll
    for (int r = 0; r < 8; ++r) { acc0[n][r] = 0.f; acc1[n][r] = 0.f; }

  const int ko = half * 2;
  for (int ks = 0; ks < U_IN / 4; ++ks) {
    const int k = ks * 4 + ko;
    v2f a0 = fetch_xu(h, mx, mn, st, vr, rb0, k, sc1a, sc2a);
    v2f a1 = fetch_xu(h, mx, mn, st, vr, rb1, k, sc1b, sc2b);
#pragma unroll
    for (int n = 0; n < 8; ++n) {
      const int nf = n * 16 + lq;
      v2f b = *(const v2f*)(Uw + (size_t)nf * U_IN + k);
      acc0[n] = wmma4(a0, b, acc0[n]);
      acc1[n] = wmma4(a1, b, acc1[n]);
    }
  }

  const float rs = rsqrtf((float)NODES);      // sqrt(1/N)
#pragma unroll
  for (int n = 0; n < 8; ++n) {
    const int f = n * 16 + lq;
    const float bias = Ub[f];
#pragma unroll
    for (int r = 0; r < 8; ++r) {
      const int node0 = n0 + r + half * 8;
      u[(size_t)node0 * DIM + f] = (acc0[n][r] + bias) * rs;
      if (t1v) {
        const int node1 = n0 + 16 + r + half * 8;
        u[(size_t)node1 * DIM + f] = (acc1[n][r] + bias) * rs;
      }
    }
  }
}

// ---------------------------------------------------------------- BN stats -> fused scale/shift
__global__ void __launch_bounds__(256) bn_stats_kernel(
    const float* __restrict__ u, const float* __restrict__ g, const float* __restrict__ b,
    float* __restrict__ bnscale, float* __restrict__ bnshift) {
  const int f = blockIdx.x;
  __shared__ float ssum[256], ssq[256];
  float s = 0.f, q = 0.f;
  for (int i = threadIdx.x; i < NODES; i += 256) {
    float v = u[(size_t)i * DIM + f];
    s += v; q += v * v;
  }
  ssum[threadIdx.x] = s; ssq[threadIdx.x] = q;
  __syncthreads();
  for (int off = 128; off > 0; off >>= 1) {
    if (threadIdx.x < off) {
      ssum[threadIdx.x] += ssum[threadIdx.x + off];
      ssq[threadIdx.x]  += ssq[threadIdx.x + off];
    }
    __syncthreads();
  }
  if (threadIdx.x == 0) {
    const float invN = 1.0f / (float)NODES;
    const float mu  = ssum[0] * invN;
    const float var = fmaxf(ssq[0] * invN - mu * mu, 0.f);
    const float sc  = g[f] * rsqrtf(var + 1e-5f);
    bnscale[f] = sc;
    bnshift[f] = b[f] - mu * sc;
  }
}

// ---------------------------------------------------------------- mix GEMM (BN fused into A-fetch) + leaky + residual
__global__ void __launch_bounds__(256) mix_gemm_kernel(
    const float* __restrict__ u, const float* __restrict__ bnscale, const float* __restrict__ bnshift,
    const float* __restrict__ Wm, const float* __restrict__ bm,
    const float* __restrict__ hin, float* __restrict__ hout) {
  const int wave = (blockIdx.x * blockDim.x + threadIdx.x) >> 5;
  const int lane = threadIdx.x & 31;
  const int half = lane >> 4;
  const int lq   = lane & 15;
  const int n0   = wave * 16;
  if (n0 >= NODES) return;
  const int rowbase = (n0 + lq) * DIM;

  v8f acc[8];
#pragma unroll
  for (int n = 0; n < 8; ++n)
#pragma unroll
    for (int r = 0; r < 8; ++r) acc[n][r] = 0.f;

  const int ko = half * 2;
  for (int ks = 0; ks < DIM / 4; ++ks) {
    const int k = ks * 4 + ko;
    v2f x  = *(const v2f*)(u + rowbase + k);
    v2f sc = *(const v2f*)(bnscale + k);
    v2f sh = *(const v2f*)(bnshift + k);
    v2f a; a.x = x.x * sc.x + sh.x; a.y = x.y * sc.y + sh.y;
#pragma unroll
    for (int n = 0; n < 8; ++n) {
      v2f b = *(const v2f*)(Wm + (size_t)(n * 16 + lq) * DIM + k);
      acc[n] = wmma4(a, b, acc[n]);
    }
  }

#pragma unroll
  for (int n = 0; n < 8; ++n) {
    const int f = n * 16 + lq;
    const float bias = bm[f];
#pragma unroll
    for (int r = 0; r < 8; ++r) {
      const int node = n0 + r + half * 8;
      float t = acc[n][r] + bias;
      t = (t > 0.f) ? t : 0.01f * t;          // leaky_relu(0.01)
      t += hin[(size_t)node * DIM + f];       // residual
      hout[(size_t)node * DIM + f] = t;
    }
  }
}

// ---------------------------------------------------------------- per-node LayerNorm + ELU (in place)
__global__ void __launch_bounds__(128) ln_elu_kernel(
    float* __restrict__ h, const float* __restrict__ g, const float* __restrict__ b) {
  const int node = blockIdx.x;
  const int f    = threadIdx.x;
  __shared__ float red[128];
  const float v = h[(size_t)node * DIM + f];
  red[f] = v; __syncthreads();
  for (int off = 64; off > 0; off >>= 1) { if (f < off) red[f] += red[f + off]; __syncthreads(); }
  const float mu = red[0] * (1.0f / DIM);
  __syncthreads();
  red[f] = (v - mu) * (v - mu); __syncthreads();
  for (int off = 64; off > 0; off >>= 1) { if (f < off) red[f] += red[f + off]; __syncthreads(); }
  const float var = red[0] * (1.0f / DIM);
  float y = (v - mu) * rsqrtf(var + 1e-5f) * g[f] + b[f];
  y = (y > 0.f) ? y : expm1f(y);              // elu
  h[(size_t)node * DIM + f] = y;
}

// ---------------------------------------------------------------- host orchestration
extern "C" void kernel_launch(void* const* d_in, const int* in_sizes, int n_in,
                              void* d_out, int out_size, void* d_ws, size_t ws_size,
                              hipStream_t stream) {
  const float* feat = (const float*)d_in[0];
  const int*   src  = (const int*)d_in[1];
  const int*   dst  = (const int*)d_in[2];
  const float* Mw   = (const float*)d_in[3];
  const float* Mb   = (const float*)d_in[4];
  const float* Uw   = (const float*)d_in[5];
  const float* Ub   = (const float*)d_in[6];
  const float* bng  = (const float*)d_in[7];
  const float* bnb  = (const float*)d_in[8];
  const float* mixw = (const float*)d_in[9];
  const float* mixb = (const float*)d_in[10];
  const float* lng  = (const float*)d_in[11];
  const float* lnb  = (const float*)d_in[12];
  float* out = (float*)d_out;

  const size_t ND = (size_t)NODES * DIM;
  float* p  = (float*)d_ws;
  float* s1 = p; p += ND;     // sum   -> std (after finalize)
  float* s2 = p; p += ND;     // sumsq -> var (after finalize)
  float* mx = p; p += ND;
  float* mn = p; p += ND;
  float* ub = p; p += ND;     // u activations
  float* hA = p; p += ND;
  float* hB = p; p += ND;
  float* dg = p; p += NODES;
  float* bnscale = p; p += DIM;
  float* bnshift = p; p += DIM;

  const int Et     = EDGES + NODES;
  const int gInit  = (NODES * DIM + 255) / 256;
  const int gDeg   = (Et + 255) / 256;
  const int wMsg   = (Et + 31) / 32;                  // one wave32 per 32-edge tile
  const int gMsg   = (wMsg * 32 + 255) / 256;
  const int wU     = (NODES + 31) / 32;               // one wave32 per 32-node tile
  const int gU     = (wU * 32 + 255) / 256;
  const int gMix   = ((NODES / 16) * 32 + 255) / 256; // one wave32 per 16-node tile

  const float* hin = feat;
  for (int l = 0; l < 3; ++l) {
    float* hout = (l == 0) ? hA : (l == 1) ? hB : out;
    const float* Mw_l = Mw + (size_t)l * DIM * TWO_DIM;
    const float* Mb_l = Mb + (size_t)l * DIM;
    const float* Uw_l = Uw + (size_t)l * DIM * U_IN;
    const float* Ub_l = Ub + (size_t)l * DIM;
    const float* bg_l = bng + (size_t)l * DIM;
    const float* bb_l = bnb + (size_t)l * DIM;
    const float* mw_l = mixw + (size_t)l * DIM * DIM;
    const float* mb_l = mixb + (size_t)l * DIM;
    const float* lg_l = lng + (size_t)l * DIM;
    const float* lb_l = lnb + (size_t)l * DIM;

    hipLaunchKernelGGL(init_accum,      dim3(gInit), dim3(256), 0, stream, s1, s2, mx, mn, dg);
    hipLaunchKernelGGL(deg_kernel,      dim3(gDeg),  dim3(256), 0, stream, dst, dg);
    hipLaunchKernelGGL(msg_kernel,      dim3(gMsg),  dim3(256), 0, stream, hin, src, dst, Mw_l, Mb_l, s1, s2, mx, mn);
    hipLaunchKernelGGL(finalize_agg,    dim3(gInit), dim3(256), 0, stream, s1, s2, dg);
    hipLaunchKernelGGL(u_gemm_kernel,   dim3(gU),    dim3(256), 0, stream, hin, mx, mn, s1, s2, dg, Uw_l, Ub_l, ub);
    hipLaunchKernelGGL(bn_stats_kernel, dim3(DIM),   dim3(256), 0, stream, ub, bg_l, bb_l, bnscale, bnshift);
    hipLaunchKernelGGL(mix_gemm_kernel, dim3(gMix),  dim3(256), 0, stream, ub, bnscale, bnshift, mw_l, mb_l, hin, hout);
    hipLaunchKernelGGL(ln_elu_kernel,   dim3(NODES), dim3(128), 0, stream, hout, lg_l, lb_l);
    hin = hout;
  }
  (void)in_sizes; (void)n_in; (void)out_size; (void)ws_size;
}